// SpMM_20968030339288
// MI455X (gfx1250) — compile-verified
//
#include <hip/hip_runtime.h>

typedef float v2f __attribute__((ext_vector_type(2)));
typedef float v8f __attribute__((ext_vector_type(8)));

#define D_FEAT 64
#define EDGES_PER_WAVE 512

// ---------------------------------------------------------------------------
// Zero d_out (harness poisons it with 0xAA before timing).
// ---------------------------------------------------------------------------
__global__ void spmm_zero_kernel(float4* __restrict__ out4, int n4) {
    int i = blockIdx.x * blockDim.x + threadIdx.x;
    if (i < n4) out4[i] = make_float4(0.f, 0.f, 0.f, 0.f);
}

// ---------------------------------------------------------------------------
// WMMA-based scatter SpMM.
// Each wave owns a contiguous chunk of EDGES_PER_WAVE edges. It keeps a
// 16-row x 64-feature fp32 tile (4x v8f accumulators) in VGPRs and feeds
// groups of 4 edges per v_wmma_f32_16x16x4_f32 step:
//   A[m,k] = (row[e+k] == base+m) ? values[e+k] : 0   (one-hot scatter)
//   B[k,n] = x[col[e+k]][n + 16*f]                    (gathered features)
// When an edge's row leaves the 16-row window the tile is flushed with
// fp32 global atomics (skipping exact zeros) and the window is rebased.
// Rows are sorted, so accepted edges in each 4-group are a prefix and the
// window advances monotonically -> ~1 flush per ~256 edges on average.
// All branches enclosing the WMMA are wave-uniform (EXEC all ones).
// ---------------------------------------------------------------------------
__global__ __launch_bounds__(256) void spmm_wmma_kernel(
    const int*   __restrict__ row,
    const int*   __restrict__ col,
    const float* __restrict__ values,
    const float* __restrict__ x,
    float*       __restrict__ out,
    int nEdges, int nNodes)
{
    const int lane  = threadIdx.x & 31;
    const int wave  = blockIdx.x * (blockDim.x >> 5) + (threadIdx.x >> 5);

    int e = wave * EDGES_PER_WAVE;
    if (e >= nEdges) return;                       // wave-uniform exit
    const int e_end = min(e + EDGES_PER_WAVE, nEdges);

    const int n16   = lane & 15;          // A row index m / B col index n / C col
    const int khalf = (lane >> 4) << 1;   // lanes 0-15 -> K{0,1}; lanes 16-31 -> K{2,3}
    const int mbase = (lane >> 4) << 3;   // C tile: rows 0-7 vs 8-15

    v8f acc0 = {}, acc1 = {}, acc2 = {}, acc3 = {};
    int base = row[e];

    auto flush = [&](int b) {
#pragma unroll
        for (int j = 0; j < 8; ++j) {
            const int rr = b + mbase + j;
            const float f0 = acc0[j], f1 = acc1[j], f2 = acc2[j], f3 = acc3[j];
            if (rr < nNodes) {
                float* orow = out + (size_t)rr * D_FEAT + n16;
                if (f0 != 0.0f) atomicAdd(orow,      f0);
                if (f1 != 0.0f) atomicAdd(orow + 16, f1);
                if (f2 != 0.0f) atomicAdd(orow + 32, f2);
                if (f3 != 0.0f) atomicAdd(orow + 48, f3);
            }
            acc0[j] = 0.0f; acc1[j] = 0.0f; acc2[j] = 0.0f; acc3[j] = 0.0f;
        }
    };

    while (e < e_end) {
        // Stream-prefetch edge metadata ahead (global_prefetch_b8).
        __builtin_prefetch((const void*)(row    + e + 128), 0, 1);
        __builtin_prefetch((const void*)(col    + e + 128), 0, 1);
        __builtin_prefetch((const void*)(values + e + 128), 0, 1);

        // All four rows of this group (same address in every lane -> cached).
        const int e0c = min(e + 0, nEdges - 1);
        const int e1c = min(e + 1, nEdges - 1);
        const int e2c = min(e + 2, nEdges - 1);
        const int e3c = min(e + 3, nEdges - 1);
        const int r0 = row[e0c], r1 = row[e1c], r2 = row[e2c], r3 = row[e3c];

        // Prefix-accept: edge k is used iff it exists in this chunk and its
        // row lies inside the current 16-row window. Rows are sorted, so
        // the accepted set is a prefix of {0,1,2,3}.
        const bool k0 =        (e + 0 < e_end) && (r0 - base < 16);
        const bool k1 = k0 && (e + 1 < e_end) && (r1 - base < 16);
        const bool k2 = k1 && (e + 2 < e_end) && (r2 - base < 16);
        const bool k3 = k2 && (e + 3 < e_end) && (r3 - base < 16);
        const int  na = (int)k0 + (int)k1 + (int)k2 + (int)k3;   // wave-uniform

        // This lane's pair of K slots.
        const int  ea = min(e + khalf,     nEdges - 1);
        const int  eb = min(e + khalf + 1, nEdges - 1);
        const int  ra = (khalf == 0) ? r0 : r2;
        const int  rb = (khalf == 0) ? r1 : r3;
        const bool oa = (khalf == 0) ? k0 : k2;
        const bool ob = (khalf == 0) ? k1 : k3;

        // A: scaled one-hot scatter matrix. Rejected edges contribute 0
        // in every lane, so their (garbage-but-in-bounds) B columns are inert.
        v2f A;
        A.x = (oa && (ra - base == n16)) ? values[ea] : 0.0f;
        A.y = (ob && (rb - base == n16)) ? values[eb] : 0.0f;

        // B: gathered feature rows; lanes 0-15 read consecutive floats of one
        // x-row per K slot -> 64B coalesced transactions.
        const float* xa = x + (size_t)col[ea] * D_FEAT;
        const float* xb = x + (size_t)col[eb] * D_FEAT;
        v2f B0, B1, B2, B3;
        B0.x = xa[n16];      B0.y = xb[n16];
        B1.x = xa[n16 + 16]; B1.y = xb[n16 + 16];
        B2.x = xa[n16 + 32]; B2.y = xb[n16 + 32];
        B3.x = xa[n16 + 48]; B3.y = xb[n16 + 48];

        // 16x16 fp32 tile update per 16-feature block.
        acc0 = __builtin_amdgcn_wmma_f32_16x16x4_f32(false, A, false, B0, (short)0, acc0, false, false);
        acc1 = __builtin_amdgcn_wmma_f32_16x16x4_f32(false, A, false, B1, (short)0, acc1, false, false);
        acc2 = __builtin_amdgcn_wmma_f32_16x16x4_f32(false, A, false, B2, (short)0, acc2, false, false);
        acc3 = __builtin_amdgcn_wmma_f32_16x16x4_f32(false, A, false, B3, (short)0, acc3, false, false);

        e += na;
        if (na < 4) {                 // window exhausted (or chunk end): flush + rebase
            flush(base);
            if (e < e_end) base = row[e];
        }
    }
    flush(base);                      // tail flush (no-op if already flushed)
}

// ---------------------------------------------------------------------------
// Harness entry point. Inputs (setup_inputs order): row[E] i32, col[E] i32,
// values[E] f32, x[N*64] f32. Output: out[N*64] f32.
// ---------------------------------------------------------------------------
extern "C" void kernel_launch(void* const* d_in, const int* in_sizes, int n_in,
                              void* d_out, int out_size, void* d_ws, size_t ws_size,
                              hipStream_t stream) {
    const int*   row    = (const int*)  d_in[0];
    const int*   col    = (const int*)  d_in[1];
    const float* values = (const float*)d_in[2];
    const float* x      = (const float*)d_in[3];
    float*       out    = (float*)      d_out;

    const int nEdges = in_sizes[0];
    const int nNodes = out_size / D_FEAT;

    // 1) zero the output (out_size = N*64 is divisible by 4)
    const int n4 = out_size / 4;
    spmm_zero_kernel<<<(n4 + 255) / 256, 256, 0, stream>>>((float4*)out, n4);

    // 2) scatter SpMM, one wave per 512-edge chunk
    const int nWaves        = (nEdges + EDGES_PER_WAVE - 1) / EDGES_PER_WAVE;
    const int wavesPerBlock = 256 / 32;
    const int nBlocks       = (nWaves + wavesPerBlock - 1) / wavesPerBlock;
    spmm_wmma_kernel<<<nBlocks, 256, 0, stream>>>(row, col, values, x, out,
                                                  nEdges, nNodes);
}